// YOLOLoss_11063835754778
// MI455X (gfx1250) — compile-verified
//
#include <hip/hip_runtime.h>

#define S_DIM 7
#define B_BOX 2
#define C_CLS 20
#define CELL_F 30            // B*5 + C
#define LAMBDA_COORD 5.0f
#define LAMBDA_NOOBJ 0.5f
#define EPSF 1e-6f

typedef float v2f __attribute__((ext_vector_type(2)));
typedef float v4f __attribute__((ext_vector_type(4)));
typedef float v8f __attribute__((ext_vector_type(8)));

__device__ __forceinline__ float sq(float x) { return x * x; }

// IoU of pred box (x,y,w,h) vs target box (tx,ty,tw,th), clamped like
// jnp.where(iou > 0, iou, 0)  (NaN > 0 is false -> 0, matching jnp.argmax input)
__device__ __forceinline__ float iou_clip(float x, float y, float w, float h,
                                          float tx, float ty, float tw, float th) {
    float b1x1 = x - 0.5f * w, b1x2 = x + 0.5f * w;
    float b1y1 = y - 0.5f * h, b1y2 = y + 0.5f * h;
    float b2x1 = tx - 0.5f * tw, b2x2 = tx + 0.5f * tw;
    float b2y1 = ty - 0.5f * th, b2y2 = ty + 0.5f * th;
    float iw = fmaxf(fminf(b1x2, b2x2) - fmaxf(b1x1, b2x1), 0.0f);
    float ih = fmaxf(fminf(b1y2, b2y2) - fmaxf(b1y1, b2y1), 0.0f);
    float inter = iw * ih;
    float uni = w * h + tw * th - inter;
    float iou = inter / (uni + EPSF);
    return (iou > 0.0f) ? iou : 0.0f;
}

// Full per-cell loss contribution (loss-term weights folded in; final /N later).
__device__ __forceinline__ float cell_contrib(const float* __restrict__ p,
                                              const float* __restrict__ t) {
    // Best box: argmax of clipped IoUs; first index wins ties.
    float c0 = iou_clip(p[0], p[1], p[2], p[3], t[0], t[1], t[2], t[3]);
    float c1 = iou_clip(p[5], p[6], p[7], p[8], t[0], t[1], t[2], t[3]);
    int best = (c1 > c0) ? 1 : 0;
    const float* bb = p + 5 * best;

    float obj = (t[4] > 0.0f) ? 1.0f : 0.0f;

    float xy_err = sq(bb[0] - t[0]) + sq(bb[1] - t[1]);
    float wh_err = sq(sqrtf(fmaxf(bb[2], EPSF)) - sqrtf(fmaxf(t[2], EPSF)))
                 + sq(sqrtf(fmaxf(bb[3], EPSF)) - sqrtf(fmaxf(t[3], EPSF)));
    float coord = xy_err + wh_err;

    float conf_obj = sq(bb[4] - t[4]);

    float cs0 = p[4] * p[4];
    float cs1 = p[9] * p[9];
    float cs_all = cs0 + cs1;
    float cs_best = best ? cs1 : cs0;
    float noobj = obj * (cs_all - cs_best) + (1.0f - obj) * cs_all;

    float cls = 0.0f;
#pragma unroll
    for (int j = B_BOX * 5; j < CELL_F; ++j) {
        float d = p[j] - t[j];
        cls += d * d;
    }

    return obj * (LAMBDA_COORD * coord + conf_obj + cls) + LAMBDA_NOOBJ * noobj;
}

__global__ __launch_bounds__(256)
void yolo_partial_kernel(const float* __restrict__ pred,
                         const float* __restrict__ tgt,
                         float* __restrict__ block_partials,
                         int ncells) {
    const int tid    = blockIdx.x * blockDim.x + threadIdx.x;
    const int stride = gridDim.x * blockDim.x;

    float local = 0.0f;

    // Process cells in PAIRS: 2 cells = 240 contiguous bytes, 16-byte aligned,
    // so the stream is pure global_load_b128 (non-temporal: single-pass data
    // roughly the size of L2, don't let it thrash the cache).
    const int npairs = ncells >> 1;
    for (int q = tid; q < npairs; q += stride) {
        const v4f* pp = (const v4f*)(pred + (size_t)q * (2 * CELL_F));
        const v4f* tp = (const v4f*)(tgt  + (size_t)q * (2 * CELL_F));
        float p[2 * CELL_F], t[2 * CELL_F];
#pragma unroll
        for (int j = 0; j < (2 * CELL_F) / 4; ++j) {
            v4f a = __builtin_nontemporal_load(pp + j);
            v4f b = __builtin_nontemporal_load(tp + j);
            p[4 * j] = a.x; p[4 * j + 1] = a.y; p[4 * j + 2] = a.z; p[4 * j + 3] = a.w;
            t[4 * j] = b.x; t[4 * j + 1] = b.y; t[4 * j + 2] = b.z; t[4 * j + 3] = b.w;
        }
        local += cell_contrib(p, t) + cell_contrib(p + CELL_F, t + CELL_F);
    }

    // Odd-cell tail (dead for N*S*S even; reconverges before the WMMA).
    if ((ncells & 1) && tid == 0) {
        const float* pp = pred + (size_t)(ncells - 1) * CELL_F;
        const float* tp = tgt  + (size_t)(ncells - 1) * CELL_F;
        float p[CELL_F], t[CELL_F];
#pragma unroll
        for (int j = 0; j < CELL_F; ++j) { p[j] = pp[j]; t[j] = tp[j]; }
        local += cell_contrib(p, t);
    }

    // ---- Wave32 reduction on the matrix unit (full f32 precision) ----
    // D = A(16x4, all ones) x B(4x16) + 0  ->  every row of D holds the 16
    // column sums of B. Each lane deposits its partial into exactly one B slot
    // (B VGPR0 = local, B VGPR1 = 0), so sum over the 16 column sums == sum of
    // all 32 lane partials, independent of the exact (k,n) slot mapping.
    // Executed after reconvergence: EXEC is all ones as WMMA requires.
    v2f A;  A.x = 1.0f;  A.y = 1.0f;
    v2f Bv; Bv.x = local; Bv.y = 0.0f;
    v8f Cz = {};
    v8f D = __builtin_amdgcn_wmma_f32_16x16x4_f32(
        /*neg_a=*/false, A, /*neg_b=*/false, Bv,
        /*c_mod=*/(short)0, Cz, /*reuse_a=*/false, /*reuse_b=*/false);

    // D VGPR0: lane L holds colsum[L % 16] (rows of D are identical).
    // Summing over lanes differing in bits 0..3 yields the full wave total.
    float s = D[0];
    s += __shfl_xor(s, 1, 32);
    s += __shfl_xor(s, 2, 32);
    s += __shfl_xor(s, 4, 32);
    s += __shfl_xor(s, 8, 32);

    __shared__ float wave_sums[8];
    const int lane = threadIdx.x & 31;
    const int wave = threadIdx.x >> 5;
    if (lane == 0) wave_sums[wave] = s;
    __syncthreads();
    if (threadIdx.x == 0) {
        float b = 0.0f;
#pragma unroll
        for (int w = 0; w < 8; ++w) b += wave_sums[w];
        block_partials[blockIdx.x] = b;
    }
}

// Deterministic final reduction: fixed-order tree over block partials.
__global__ __launch_bounds__(256)
void yolo_final_kernel(const float* __restrict__ block_partials,
                       float* __restrict__ out, int nblocks, float inv_n) {
    __shared__ float sm[256];
    float a = 0.0f;
    for (int i = threadIdx.x; i < nblocks; i += 256) a += block_partials[i];
    sm[threadIdx.x] = a;
    __syncthreads();
#pragma unroll
    for (int off = 128; off > 0; off >>= 1) {
        if (threadIdx.x < off) sm[threadIdx.x] += sm[threadIdx.x + off];
        __syncthreads();
    }
    if (threadIdx.x == 0) out[0] = sm[0] * inv_n;
}

extern "C" void kernel_launch(void* const* d_in, const int* in_sizes, int n_in,
                              void* d_out, int out_size, void* d_ws, size_t ws_size,
                              hipStream_t stream) {
    (void)n_in; (void)out_size;
    const float* pred = (const float*)d_in[0];
    const float* tgt  = (const float*)d_in[1];
    float* out = (float*)d_out;
    float* ws  = (float*)d_ws;

    const int ncells = in_sizes[0] / CELL_F;          // N * S * S
    const int nN     = ncells / (S_DIM * S_DIM);      // N
    const float inv_n = 1.0f / (float)nN;

    const int threads = 256;                          // 8 wave32s / block
    const int npairs  = ncells >> 1;
    int blocks = (npairs + threads - 1) / threads;
    if (blocks > 1024) blocks = 1024;                 // ~3 cells/thread: amortize epilogue
    if (blocks < 1) blocks = 1;
    const int ws_cap = (int)(ws_size / sizeof(float));
    if (ws_cap > 0 && blocks > ws_cap) blocks = ws_cap;

    yolo_partial_kernel<<<blocks, threads, 0, stream>>>(pred, tgt, ws, ncells);
    yolo_final_kernel<<<1, 256, 0, stream>>>(ws, out, blocks, inv_n);
}